// FAE_SAGEConv_77653008712165
// MI455X (gfx1250) — compile-verified
//
#include <hip/hip_runtime.h>

#define NN 100000
#define NE 1600000

typedef __attribute__((ext_vector_type(2))) float v2f;
typedef __attribute__((ext_vector_type(8))) float v8f;

// ---------------------------------------------------------------- zero ws
__global__ void FAE_zero_kernel(float* __restrict__ p, long long n) {
    long long i = (long long)blockIdx.x * blockDim.x + threadIdx.x;
    long long stride = (long long)gridDim.x * blockDim.x;
    for (; i < n; i += stride) p[i] = 0.0f;
}

// ------------------------------------------------- layer-1 scatter (50 ch)
// thread = (edge, channel), channel = idx & 63: lanes of a wave read
// consecutive channels of one source row (coalesced) and atomically add to
// consecutive dest channels. x (20MB) and agg1 (20MB) are L2-resident.
__global__ void FAE_agg1_kernel(const float* __restrict__ x,
                                const int* __restrict__ src,
                                const int* __restrict__ dst,
                                float* __restrict__ agg,
                                float* __restrict__ cnt) {
    long long idx = (long long)blockIdx.x * blockDim.x + threadIdx.x;
    if (idx >= (long long)NE * 64) return;
    int e = (int)(idx >> 6);
    int c = (int)(idx & 63);
    int d = dst[e];
    if (c < 50) {
        int s = src[e];
        atomicAdd(&agg[(long long)d * 50 + c], x[(long long)s * 50 + c]);
    } else if (c == 63) {
        atomicAdd(&cnt[d], 1.0f);
    }
}

// ------------------------------------------------- layer-2 scatter (64 ch)
__global__ void FAE_agg2_kernel(const float* __restrict__ h1,
                                const int* __restrict__ src,
                                const int* __restrict__ dst,
                                float* __restrict__ agg) {
    long long idx = (long long)blockIdx.x * blockDim.x + threadIdx.x;
    if (idx >= (long long)NE * 64) return;
    int e = (int)(idx >> 6);
    int c = (int)(idx & 63);
    atomicAdd(&agg[(long long)dst[e] * 64 + c], h1[(long long)src[e] * 64 + c]);
}

// ---------------------------------------------------------------- GEMM 1
// h1[16-row tile] = relu(concat(x, agg1/deg) @ W1 + b1), K=100, N=64.
// One wave per tile; f32 WMMA 16x16x4 per ISA VGPR layouts:
//   A: lane(0..15)=row M, vgpr pair = K{k0,k0+1}; lanes 16..31 = K{k0+2,k0+3}
//   B: lane%16 = col N, same K split by half-wave
//   C/D: vgpr j = rows {j, j+8}, N = lane%16
// K loop split at the concat boundary (50) so inner loops are branch-free.
__global__ __launch_bounds__(256)
void FAE_gemm1_kernel(const float* __restrict__ x,
                      const float* __restrict__ agg1,
                      const float* __restrict__ cnt,
                      const float* __restrict__ W1,
                      const float* __restrict__ b1,
                      float* __restrict__ h1) {
    const int lane = threadIdx.x & 31;
    const int wave = threadIdx.x >> 5;
    const int tile = blockIdx.x * 8 + wave;
    if (tile * 16 >= NN) return;                 // wave-uniform exit (EXEC all-1 for WMMA)
    const int r    = lane & 15;
    const int half = lane >> 4;
    const int m    = tile * 16 + r;

    const float inv = 1.0f / fmaxf(cnt[m], 1.0f);
    const float* __restrict__ xrow = x    + (long long)m * 50;
    const float* __restrict__ arow = agg1 + (long long)m * 50;

    v8f acc0 = {}, acc1 = {}, acc2 = {}, acc3 = {};

    #define FAE_G1_STEP(AX, AY, KA, KB)                                                              \
        {                                                                                            \
            v2f a; a.x = (AX); a.y = (AY);                                                           \
            const float* __restrict__ wa = W1 + (long long)(KA) * 64;                                \
            const float* __restrict__ wb = W1 + (long long)(KB) * 64;                                \
            v2f b;                                                                                   \
            b.x = wa[r];      b.y = wb[r];                                                           \
            acc0 = __builtin_amdgcn_wmma_f32_16x16x4_f32(false, a, false, b, (short)0, acc0, false, false); \
            b.x = wa[16 + r]; b.y = wb[16 + r];                                                      \
            acc1 = __builtin_amdgcn_wmma_f32_16x16x4_f32(false, a, false, b, (short)0, acc1, false, false); \
            b.x = wa[32 + r]; b.y = wb[32 + r];                                                      \
            acc2 = __builtin_amdgcn_wmma_f32_16x16x4_f32(false, a, false, b, (short)0, acc2, false, false); \
            b.x = wa[48 + r]; b.y = wb[48 + r];                                                      \
            acc3 = __builtin_amdgcn_wmma_f32_16x16x4_f32(false, a, false, b, (short)0, acc3, false, false); \
        }

    // K chunks 0..44: both lane halves read pure x (K <= 47)
    for (int k0 = 0; k0 < 48; k0 += 4) {
        const int ka = k0 + 2 * half;
        FAE_G1_STEP(xrow[ka], xrow[ka + 1], ka, ka + 1);
    }
    // Straddle chunk k0=48: half0 -> x[48..49], half1 -> agg[0..1]*inv.
    // Per-lane pointer+scale select (v_cndmask), no EXEC divergence.
    {
        const float* __restrict__ p = half ? arow : (xrow + 48);
        const float s = half ? inv : 1.0f;
        const int ka = 48 + 2 * half;
        FAE_G1_STEP(p[0] * s, p[1] * s, ka, ka + 1);
    }
    // K chunks 52..96: both lane halves read pure agg (K >= 52)
    for (int k0 = 52; k0 < 100; k0 += 4) {
        const int ka = k0 + 2 * half;
        FAE_G1_STEP(arow[ka - 50] * inv, arow[ka - 49] * inv, ka, ka + 1);
    }
    #undef FAE_G1_STEP

    const float bn0 = b1[r], bn1 = b1[16 + r], bn2 = b1[32 + r], bn3 = b1[48 + r];
    const int mbase = tile * 16 + half * 8;
    #pragma unroll
    for (int j = 0; j < 8; ++j) {
        float* __restrict__ o = h1 + (long long)(mbase + j) * 64;
        o[r]      = fmaxf(acc0[j] + bn0, 0.0f);
        o[16 + r] = fmaxf(acc1[j] + bn1, 0.0f);
        o[32 + r] = fmaxf(acc2[j] + bn2, 0.0f);
        o[48 + r] = fmaxf(acc3[j] + bn3, 0.0f);
    }
}

// ---------------------------------------------------------------- GEMM 2 + head
// h2 = relu(concat(h1, agg2/deg) @ W2 + b2) kept in registers; out = h2 @ W3 + b3
// fused via shfl_xor reduction over each 16-lane half (row m's 32 cols live there).
// Concat boundary (64) is chunk-aligned -> two fully branch-free K loops.
__global__ __launch_bounds__(256)
void FAE_gemm2_kernel(const float* __restrict__ h1,
                      const float* __restrict__ agg2,
                      const float* __restrict__ cnt,
                      const float* __restrict__ W2,
                      const float* __restrict__ b2,
                      const float* __restrict__ W3,
                      const float* __restrict__ b3,
                      float* __restrict__ out) {
    const int lane = threadIdx.x & 31;
    const int wave = threadIdx.x >> 5;
    const int tile = blockIdx.x * 8 + wave;
    if (tile * 16 >= NN) return;                 // wave-uniform
    const int r    = lane & 15;
    const int half = lane >> 4;
    const int m    = tile * 16 + r;

    const float inv = 1.0f / fmaxf(cnt[m], 1.0f);
    const float* __restrict__ hrow = h1   + (long long)m * 64;
    const float* __restrict__ arow = agg2 + (long long)m * 64;

    v8f acc0 = {}, acc1 = {};

    #define FAE_G2_STEP(AX, AY, KA, KB)                                                              \
        {                                                                                            \
            v2f a; a.x = (AX); a.y = (AY);                                                           \
            const float* __restrict__ wa = W2 + (long long)(KA) * 32;                                \
            const float* __restrict__ wb = W2 + (long long)(KB) * 32;                                \
            v2f b;                                                                                   \
            b.x = wa[r];      b.y = wb[r];                                                           \
            acc0 = __builtin_amdgcn_wmma_f32_16x16x4_f32(false, a, false, b, (short)0, acc0, false, false); \
            b.x = wa[16 + r]; b.y = wb[16 + r];                                                      \
            acc1 = __builtin_amdgcn_wmma_f32_16x16x4_f32(false, a, false, b, (short)0, acc1, false, false); \
        }

    // K 0..63: h1 part of concat
    for (int k0 = 0; k0 < 64; k0 += 4) {
        const int ka = k0 + 2 * half;
        FAE_G2_STEP(hrow[ka], hrow[ka + 1], ka, ka + 1);
    }
    // K 64..127: mean part of concat
    for (int k0 = 64; k0 < 128; k0 += 4) {
        const int ka = k0 + 2 * half;
        FAE_G2_STEP(arow[ka - 64] * inv, arow[ka - 63] * inv, ka, ka + 1);
    }
    #undef FAE_G2_STEP

    const float bias0 = b2[r], bias1 = b2[16 + r];
    const float w3a = W3[r], w3b = W3[16 + r];
    const float bb3 = b3[0];
    const int mbase = tile * 16 + half * 8;
    #pragma unroll
    for (int j = 0; j < 8; ++j) {
        float v0 = fmaxf(acc0[j] + bias0, 0.0f);     // (row mbase+j, col r)
        float v1 = fmaxf(acc1[j] + bias1, 0.0f);     // (row mbase+j, col 16+r)
        float t = v0 * w3a + v1 * w3b;
        t += __shfl_xor(t, 1, 32);
        t += __shfl_xor(t, 2, 32);
        t += __shfl_xor(t, 4, 32);
        t += __shfl_xor(t, 8, 32);                   // sum over the 16-lane half
        if (r == 0) out[mbase + j] = t + bb3;
    }
}

// ---------------------------------------------------------------- launch
extern "C" void kernel_launch(void* const* d_in, const int* in_sizes, int n_in,
                              void* d_out, int out_size, void* d_ws, size_t ws_size,
                              hipStream_t stream) {
    const float* x  = (const float*)d_in[0];
    const int*   ei = (const int*)d_in[1];
    const float* W1 = (const float*)d_in[2];
    const float* b1 = (const float*)d_in[3];
    const float* W2 = (const float*)d_in[4];
    const float* b2 = (const float*)d_in[5];
    const float* W3 = (const float*)d_in[6];
    const float* b3 = (const float*)d_in[7];
    float* out = (float*)d_out;

    float* ws   = (float*)d_ws;
    float* agg1 = ws;               //  5,000,000 f32
    float* cnt  = ws + 5000000;     //    100,000 f32
    float* h1   = ws + 5100000;     //  6,400,000 f32
    float* agg2 = ws + 11500000;    //  6,400,000 f32  (total 71.6 MB)

    const int* src = ei;            // edge_index[0]
    const int* dst = ei + NE;       // edge_index[1]

    const long long zn = 17900000;  // zero everything we accumulate into
    FAE_zero_kernel<<<4096, 256, 0, stream>>>(ws, zn);

    const long long ecthreads = (long long)NE * 64;
    const int eblocks = (int)((ecthreads + 255) / 256);
    FAE_agg1_kernel<<<eblocks, 256, 0, stream>>>(x, src, dst, agg1, cnt);

    const int gblocks = (NN / 16 + 7) / 8;   // 8 waves/block, 1 tile/wave
    FAE_gemm1_kernel<<<gblocks, 256, 0, stream>>>(x, agg1, cnt, W1, b1, h1);

    FAE_agg2_kernel<<<eblocks, 256, 0, stream>>>(h1, src, dst, agg2);

    FAE_gemm2_kernel<<<gblocks, 256, 0, stream>>>(h1, agg2, cnt, W2, b2, W3, b3, out);
}